// DDDM_88837103551016
// MI455X (gfx1250) — compile-verified
//
#include <hip/hip_runtime.h>

// CDNA5 / gfx1250: fourier-feature + polynomial evaluation as a skinny GEMM
// [N,40] x [40,4] using V_WMMA_F32_16X16X4_F32 (full fp32 precision).
// Memory-bound: 192 B/point, ~1.7 FLOP/B -> HBM roofline ~17 us at 23.3 TB/s.

typedef __attribute__((ext_vector_type(2))) float v2f;
typedef __attribute__((ext_vector_type(4))) float v4f;
typedef __attribute__((ext_vector_type(8))) float v8f;

#define KH 18            // harmonics (LOW+MID+HIGH = 3+6+9)
#define NPOLY 4          // POLY_DEGREE + 1
#define BATCH 4
#define TWO_PI 6.28318530717958647692f

__global__ __launch_bounds__(256) void fourier_poly_wmma(
    const float* __restrict__ t_in,       // [4]
    const float* __restrict__ poly,       // [N,4]
    const float* __restrict__ fa,         // [N,18]
    const float* __restrict__ fb,         // [N,18]
    const int*   __restrict__ stage_in,   // [1]
    float* __restrict__ out,              // [2, B, N] (y_poly then y_fourier)
    int npts)
{
  const int lane = threadIdx.x & 31;
  const int col  = lane & 15;   // WMMA N column (batch index when < 4)
  const int half = lane >> 4;   // 0 -> K{0,1}, 1 -> K{2,3} of each K=4 chunk
  const bool live = (col < BATCH);

  // ---- stage gating (model_stage input), matches reference clamping ----
  int ms   = stage_in[0];
  int curr = (ms >= 0) ? (ms < 3 ? ms : 3) : 3;

  // ---- per-wave constant B-operand tables (computed once) ----
  const float t  = live ? t_in[col] : 0.0f;
  const float t2 = t * t, t3 = t2 * t;
  const float tp[4] = {1.0f, t, t2, t3};

  v2f bpoly;
  bpoly[0] = live ? tp[2 * half + 0] : 0.0f;
  bpoly[1] = live ? tp[2 * half + 1] : 0.0f;

  // c[k] = a[k] (k<18) else b[k-18]; weights w[k] = cos/sin(2*pi*(k+1)*t),
  // gated by band-vs-stage, zeroed for dead columns.
  v2f bfv[9];
#pragma unroll
  for (int j = 0; j < 9; ++j) {
#pragma unroll
    for (int e = 0; e < 2; ++e) {
      const int kk = 4 * j + 2 * half + e;               // 0..35
      const int k  = (kk < KH) ? kk : (kk - KH);          // harmonic index
      const int need = (k < 3) ? 1 : ((k < 9) ? 2 : 3);   // band's req stage
      const float g  = (live && curr >= need) ? 1.0f : 0.0f;
      const float ph = TWO_PI * (float)(k + 1) * t;
      const float wv = (kk < KH) ? cosf(ph) : sinf(ph);
      bfv[j][e] = g * wv;
    }
  }

  // ---- grid-stride over 16-point tiles ----
  const int tiles     = npts >> 4;
  const int wavesPerB = blockDim.x >> 5;
  const int wid       = blockIdx.x * wavesPerB + (threadIdx.x >> 5);
  const int nwaves    = gridDim.x * wavesPerB;

  float* __restrict__ out_poly    = out;
  float* __restrict__ out_fourier = out + (size_t)BATCH * (size_t)npts;

  for (int tile = wid; tile < tiles; tile += nwaves) {
    const size_t base = (size_t)tile << 4;
    const size_t p    = base + (size_t)col;   // this lane's point row (M)

    // prefetch next tile's streams (global_prefetch_b8)
    const int nt = tile + nwaves;
    if (nt < tiles) {
      const size_t np = ((size_t)nt << 4) + (size_t)col;
      __builtin_prefetch(poly + np * NPOLY, 0, 1);
      __builtin_prefetch(fa   + np * KH,    0, 1);
      __builtin_prefetch(fb   + np * KH,    0, 1);
    }

    // ---- y_poly: one 16x16x4 f32 WMMA ----
    const v2f apoly = *(const v2f*)(poly + p * NPOLY + 2 * half);
    v8f zc = {};
    v8f accp = __builtin_amdgcn_wmma_f32_16x16x4_f32(
        false, apoly, false, bpoly, (short)0, zc, false, false);

    // ---- y_fourier: nine 16x16x4 f32 WMMAs over concat(a,b) (K=36) ----
    v8f accf = {};
#pragma unroll
    for (int j = 0; j < 9; ++j) {
      const int k0 = 4 * j + 2 * half;
      const float* src = (k0 < KH) ? (fa + p * KH + k0)
                                   : (fb + p * KH + (k0 - KH));
      const v2f af = *(const v2f*)src;
      accf = __builtin_amdgcn_wmma_f32_16x16x4_f32(
          false, af, false, bfv[j], (short)0, accf, false, false);
    }

    // ---- store: lane holds D[M = 8*half + r][N = col], r = 0..7 ->
    // 8 consecutive points for batch `col`; two aligned float4 stores each.
    if (live) {
      const size_t o = (size_t)col * (size_t)npts + base + (size_t)(half * 8);
      v4f plo = {accp[0], accp[1], accp[2], accp[3]};
      v4f phi = {accp[4], accp[5], accp[6], accp[7]};
      v4f flo = {accf[0], accf[1], accf[2], accf[3]};
      v4f fhi = {accf[4], accf[5], accf[6], accf[7]};
      *(v4f*)(out_poly    + o)     = plo;
      *(v4f*)(out_poly    + o + 4) = phi;
      *(v4f*)(out_fourier + o)     = flo;
      *(v4f*)(out_fourier + o + 4) = fhi;
    }
  }
}

extern "C" void kernel_launch(void* const* d_in, const int* in_sizes, int n_in,
                              void* d_out, int out_size, void* d_ws, size_t ws_size,
                              hipStream_t stream) {
  const float* t_in  = (const float*)d_in[0];
  const float* poly  = (const float*)d_in[1];
  const float* fa    = (const float*)d_in[2];
  const float* fb    = (const float*)d_in[3];
  const int*   stage = (const int*)d_in[4];
  float* out = (float*)d_out;

  const int npts = in_sizes[1] / NPOLY;   // 128^3

  const int threads = 256;                 // 8 waves (wave32)
  const int blocks  = 1024;                // 8192 waves, ~16 tiles each
  fourier_poly_wmma<<<blocks, threads, 0, stream>>>(
      t_in, poly, fa, fb, stage, out, npts);
}